// Attention_5781025980827
// MI455X (gfx1250) — compile-verified
//
#include <hip/hip_runtime.h>
#include <math.h>

// Problem: B=32, L=2048, D=1024, f32. Bandwidth-bound (256MB enc, 0.27 GFLOP).
// Single HBM pass: online softmax, async-DMA LDS staging, WMMA f32 16x16x4
// for the weighted context accumulation. decoder_hidden / bias are
// softmax-shift-invariant and therefore ignored.

#define BATCH   32
#define SEQL    2048
#define DIM     1024
#define NCHUNK  8                    // L-chunks per batch -> 256 blocks
#define RCHUNK  (SEQL / NCHUNK)      // 256 rows per chunk
#define TR      16                   // rows per LDS tile
#define NTILE   (RCHUNK / TR)        // 16 tiles per chunk
#define ROWDW   1032                 // padded LDS row stride (dwords) -> no bank conflicts
#define ROWBYTES (ROWDW * 4)         // 4128 B
#define TILEBYTES (TR * ROWBYTES)    // 66048 B per tile buffer
#define PART    1032                 // floats per (batch,chunk) partial record
#define THREADS 256                  // 8 wave32 waves

typedef float v2f __attribute__((ext_vector_type(2)));
typedef float v8f __attribute__((ext_vector_type(8)));

extern __shared__ char dynLds[];

// Stage one TR x 1024 f32 tile (contiguous in global) into LDS with padded
// rows, using CDNA5 async global->LDS DMA. Each wave issues 16 b128 copies
// (32 lanes x 16B = 512B per instruction); tile = 4096 units total.
__device__ __forceinline__ void stage_tile(unsigned ldsBase, const char* gbase,
                                           int wave, int lane) {
#pragma unroll
  for (int i = 0; i < 16; ++i) {
    unsigned u   = (unsigned)(wave * 512 + i * 32 + lane); // 16B unit index
    unsigned row = u >> 8;                                 // 256 units per row
    unsigned cu  = u & 255u;
    unsigned lds = ldsBase + row * ROWBYTES + cu * 16u;
    unsigned off = u * 16u;                                // linear global offset
    asm volatile("global_load_async_to_lds_b128 %0, %1, %2"
                 :: "v"(lds), "v"(off), "s"(gbase)
                 : "memory");
  }
}

__global__ void __launch_bounds__(THREADS)
attn_partial(const float* __restrict__ enc, const float* __restrict__ W,
             float* __restrict__ ws) {
  const int tid  = threadIdx.x;
  const int lane = tid & 31;
  const int wave = tid >> 5;
  const int bid  = blockIdx.x;
  const int b    = bid >> 3;            // batch
  const int c    = bid & (NCHUNK - 1);  // L-chunk

  // Low 32 bits of a generic LDS pointer == LDS byte offset (aperture form).
  const unsigned ldsBase = (unsigned)(size_t)(void*)dynLds;
  float* wLds   = (float*)(dynLds + 2 * TILEBYTES);
  float* logLds = (float*)(dynLds + 2 * TILEBYTES + 4096);

  // Stage w_enc (first DIM floats of W) into LDS: 256 threads x float4.
  ((float4*)wLds)[tid] = ((const float4*)W)[tid];

  const char* chunk = (const char*)enc +
      ((size_t)b * SEQL + (size_t)c * RCHUNK) * (size_t)DIM * 4u;

  stage_tile(ldsBase, chunk, wave, lane);   // prefetch tile 0 into buffer 0

  float m_run = -INFINITY, s_run = 0.0f;
  v8f C[8] = {};                            // 8 n-tiles x 16 ctx dims per wave

  const int kSel = (lane >= 16) ? 2 : 0;    // A/B K-half per ISA 16x4 layout
  const int col  = lane & 15;

  for (int t = 0; t < NTILE; ++t) {
    if (t + 1 < NTILE) {
      stage_tile(ldsBase + (unsigned)(((t + 1) & 1) * TILEBYTES),
                 chunk + (size_t)(t + 1) * TR * DIM * 4, wave, lane);
      asm volatile("s_wait_asynccnt 16" ::: "memory");  // tile t's 16 done
    } else {
      asm volatile("s_wait_asynccnt 0" ::: "memory");
    }
    __syncthreads();

    const float* tile = (const float*)(dynLds + (t & 1) * TILEBYTES);

    // ---- phase A: logits for this tile (2 rows per wave) ----
#pragma unroll
    for (int rr = 0; rr < TR / 8; ++rr) {
      const int r = wave * (TR / 8) + rr;
      const float4* row4 = (const float4*)(tile + r * ROWDW);
      const float4* w4   = (const float4*)wLds;
      float acc = 0.0f;
#pragma unroll
      for (int q = 0; q < 8; ++q) {         // lane-major: conflict-free b128
        float4 a = row4[q * 32 + lane];
        float4 w = w4[q * 32 + lane];
        acc += a.x * w.x + a.y * w.y + a.z * w.z + a.w * w.w;
      }
#pragma unroll
      for (int d = 16; d > 0; d >>= 1) acc += __shfl_xor(acc, d, 32);
      if (lane == 0) logLds[r] = acc;
    }
    __syncthreads();

    // ---- online softmax state update (uniform, computed redundantly) ----
    float tmax = -INFINITY;
#pragma unroll
    for (int r = 0; r < TR; ++r) tmax = fmaxf(tmax, logLds[r]);
    const float new_m = fmaxf(m_run, tmax);
    const float scale = __expf(m_run - new_m);  // 0 on first tile
    float ssum = 0.0f;
#pragma unroll
    for (int r = 0; r < TR; ++r) ssum += __expf(logLds[r] - new_m);
    s_run = s_run * scale + ssum;
    m_run = new_m;
#pragma unroll
    for (int j = 0; j < 8; ++j)
#pragma unroll
      for (int e = 0; e < 8; ++e) C[j][e] *= scale;

    // ---- phase B: ctx += exp(logit-m) * enc_row via WMMA f32 16x16x4 ----
    // A[m][k] = weight(row k0+k), replicated over M (all lanes in a half
    // carry the same K values per the ISA 16x4 A layout).
#pragma unroll
    for (int ks = 0; ks < TR / 4; ++ks) {
      const int k0 = ks * 4;
      v2f A;
      A[0] = __expf(logLds[k0 + kSel]     - new_m);
      A[1] = __expf(logLds[k0 + kSel + 1] - new_m);
      const float* rx = tile + (k0 + kSel) * ROWDW;       // B rows K0/K2
      const float* ry = rx + ROWDW;                       // B rows K1/K3
#pragma unroll
      for (int j = 0; j < 8; ++j) {
        const int n0 = wave * 128 + j * 16;
        v2f Bv;
        Bv[0] = rx[n0 + col];
        Bv[1] = ry[n0 + col];
        C[j] = __builtin_amdgcn_wmma_f32_16x16x4_f32(
            false, A, false, Bv, (short)0, C[j], false, false);
      }
    }
    __syncthreads();   // buffer t&1 fully consumed before reuse
  }

  // ---- write partial record: ctx (row M=0 = lanes 0..15, VGPR 0), m, s ----
  float* rec = ws + (size_t)bid * PART;
#pragma unroll
  for (int j = 0; j < 8; ++j) {
    const int n0 = wave * 128 + j * 16;
    if (lane < 16) rec[n0 + lane] = C[j][0];
  }
  if (tid == 0) { rec[DIM] = m_run; rec[DIM + 1] = s_run; }
}

// Combine NCHUNK partials per batch: ctx = sum_i ctx_i*e^(m_i-M) / S.
__global__ void __launch_bounds__(THREADS)
attn_combine(const float* __restrict__ ws, float* __restrict__ out) {
  const int b   = blockIdx.x;
  const int tid = threadIdx.x;
  float m[NCHUNK], s[NCHUNK], e[NCHUNK];
  float M = -INFINITY;
#pragma unroll
  for (int i = 0; i < NCHUNK; ++i) {
    m[i] = ws[((size_t)b * NCHUNK + i) * PART + DIM];
    s[i] = ws[((size_t)b * NCHUNK + i) * PART + DIM + 1];
    M = fmaxf(M, m[i]);
  }
  float S = 0.0f;
#pragma unroll
  for (int i = 0; i < NCHUNK; ++i) { e[i] = __expf(m[i] - M); S += s[i] * e[i]; }
  const float inv = 1.0f / S;
  for (int d = tid * 4; d < DIM; d += THREADS * 4) {
    float4 acc = {0.0f, 0.0f, 0.0f, 0.0f};
#pragma unroll
    for (int i = 0; i < NCHUNK; ++i) {
      const float4 p = *(const float4*)&ws[((size_t)b * NCHUNK + i) * PART + d];
      acc.x += p.x * e[i]; acc.y += p.y * e[i];
      acc.z += p.z * e[i]; acc.w += p.w * e[i];
    }
    acc.x *= inv; acc.y *= inv; acc.z *= inv; acc.w *= inv;
    *(float4*)&out[(size_t)b * DIM + d] = acc;
  }
}

extern "C" void kernel_launch(void* const* d_in, const int* in_sizes, int n_in,
                              void* d_out, int out_size, void* d_ws, size_t ws_size,
                              hipStream_t stream) {
  const float* enc = (const float*)d_in[0];   // encoder_output (32,2048,1024)
  // d_in[1] = decoder_hidden, d_in[3] = bias: softmax-shift-invariant, unused.
  const float* W   = (const float*)d_in[2];   // (2048,1); first 1024 = w_enc
  float* out = (float*)d_out;
  float* ws  = (float*)d_ws;                  // needs 256*1032*4 ~ 1.01 MB

  const size_t shmem = 2 * TILEBYTES + 4096 + 64;  // 136,256 B (<320 KB WGP LDS)
  attn_partial<<<dim3(BATCH * NCHUNK), dim3(THREADS), shmem, stream>>>(enc, W, ws);
  attn_combine<<<dim3(BATCH), dim3(THREADS), 0, stream>>>(ws, out);
}